// GAT_28200755265750
// MI455X (gfx1250) — compile-verified
//
#include <hip/hip_runtime.h>
#include <hip/hip_bf16.h>
#include <math.h>

typedef __attribute__((ext_vector_type(2))) float v2f;
typedef __attribute__((ext_vector_type(8))) float v8f;

#define NNODES 100000
#define NEDGES 1600000
#define ETOT   (NEDGES + NNODES)
#define DHID   64
#define NEG_SLOPE 0.2f

// ---------------------------------------------------------------------------
// Float atomic max via sign-split int/uint atomics (works with -inf init).
// ---------------------------------------------------------------------------
__device__ __forceinline__ void atomicMaxF(float* addr, float value) {
    if (value >= 0.0f)
        atomicMax((int*)addr, __float_as_int(value));
    else
        atomicMin((unsigned int*)addr, __float_as_uint(value));
}

__device__ __forceinline__ void edge_endpoints(const int* __restrict__ ei,
                                               int e, int& s, int& d) {
    if (e < NEDGES) { s = ei[e]; d = ei[NEDGES + e]; }
    else            { s = e - NEDGES; d = e - NEDGES; }
}

// ---------------------------------------------------------------------------
// h[n, dout] = x[n, dk] @ W[dk, dout]   (fp32 WMMA 16x16x4, one wave per tile)
//
// A-frag (16x4 f32, ISA 7.12.2): lanes 0-15 row M=lane, V0=K0,V1=K1;
//                                lanes 16-31 row M=lane-16, V0=K2,V1=K3.
// B-frag (4x16 f32, mirrored):   lanes 0-15 col N=lane, V0=K0,V1=K1;
//                                lanes 16-31 col N=lane-16, V0=K2,V1=K3.
// C/D (16x16 f32): VGPR r -> (M=r, N=l16) lanes 0-15, (M=r+8, N=l16) lanes 16-31.
// ---------------------------------------------------------------------------
__global__ __launch_bounds__(256) void gat_gemm_wmma_f32(
    const float* __restrict__ x, const float* __restrict__ W,
    float* __restrict__ h, int n, int dk, int dout) {
    const int wave = threadIdx.x >> 5;
    const int lane = threadIdx.x & 31;
    const int half = lane >> 4;
    const int l16  = lane & 15;
    const int colTiles = dout >> 4;
    const int task = blockIdx.x * 8 + wave;
    const int rowTile = task / colTiles;
    const int colTile = task % colTiles;
    if (rowTile * 16 >= n) return;   // wave-uniform: EXEC all-1s inside
    const int row0 = rowTile * 16;
    const int col0 = colTile * 16;

    v8f acc = {};
    const float* xrow = x + (size_t)(row0 + l16) * dk;
    for (int k = 0; k < dk; k += 4) {
        v2f a, b;
        a.x = xrow[k + 2 * half + 0];
        a.y = xrow[k + 2 * half + 1];
        b.x = W[(size_t)(k + 2 * half + 0) * dout + col0 + l16];
        b.y = W[(size_t)(k + 2 * half + 1) * dout + col0 + l16];
        acc = __builtin_amdgcn_wmma_f32_16x16x4_f32(
            /*neg_a=*/false, a, /*neg_b=*/false, b,
            /*c_mod=*/(short)0, acc, /*reuse_a=*/false, /*reuse_b=*/false);
    }
#pragma unroll
    for (int r = 0; r < 8; ++r)
        h[(size_t)(row0 + r + 8 * half) * dout + col0 + l16] = acc[r];
}

// ---------------------------------------------------------------------------
// Per-node attention dot products: s[i] = h[i]·a_src, d[i] = h[i]·a_dst
// ---------------------------------------------------------------------------
__global__ __launch_bounds__(256) void gat_att_dots(
    const float* __restrict__ h, const float* __restrict__ a_src,
    const float* __restrict__ a_dst, float* __restrict__ s,
    float* __restrict__ d, int n) {
    int i = blockIdx.x * blockDim.x + threadIdx.x;
    if (i >= n) return;
    const float4* hp = (const float4*)(h + (size_t)i * DHID);
    const float4* as = (const float4*)a_src;
    const float4* ad = (const float4*)a_dst;
    float ss = 0.0f, dd = 0.0f;
#pragma unroll
    for (int j = 0; j < DHID / 4; ++j) {
        float4 hv = hp[j], av = as[j], bv = ad[j];
        ss += hv.x * av.x + hv.y * av.y + hv.z * av.z + hv.w * av.w;
        dd += hv.x * bv.x + hv.y * bv.y + hv.z * bv.z + hv.w * bv.w;
    }
    s[i] = ss;
    d[i] = dd;
}

// ---------------------------------------------------------------------------
__global__ void fill_f32(float* __restrict__ p, float v, long long count) {
    long long i = (long long)blockIdx.x * blockDim.x + threadIdx.x;
    long long stride = (long long)gridDim.x * blockDim.x;
    for (; i < count; i += stride) p[i] = v;
}

// Pass A: logit = leaky_relu(s[src] + d[dst]); m[dst] = max(m[dst], logit)
__global__ __launch_bounds__(256) void gat_edge_logit_max(
    const int* __restrict__ ei, const float* __restrict__ s,
    const float* __restrict__ d, float* __restrict__ logit,
    float* __restrict__ m) {
    int e = blockIdx.x * blockDim.x + threadIdx.x;
    if (e >= ETOT) return;
    int si, di;
    edge_endpoints(ei, e, si, di);
    float lg = s[si] + d[di];
    lg = (lg > 0.0f) ? lg : NEG_SLOPE * lg;
    logit[e] = lg;
    atomicMaxF(&m[di], lg);
}

// Pass B: ew = exp(logit - m[dst]) (stored over logit); denom[dst] += ew
__global__ __launch_bounds__(256) void gat_edge_exp_sum(
    const int* __restrict__ ei, const float* __restrict__ m,
    float* __restrict__ logit_ew, float* __restrict__ denom) {
    int e = blockIdx.x * blockDim.x + threadIdx.x;
    if (e >= ETOT) return;
    int si, di;
    edge_endpoints(ei, e, si, di);
    float ev = expf(logit_ew[e] - m[di]);
    logit_ew[e] = ev;
    atomicAdd(&denom[di], ev);
}

// Pass C: one wave per edge; out[dst][c] += alpha * h[src][c], c = lane, lane+32
__global__ __launch_bounds__(256) void gat_edge_aggregate(
    const int* __restrict__ ei, const float* __restrict__ ew,
    const float* __restrict__ denom, const float* __restrict__ h,
    float* __restrict__ out) {
    int gwave  = (int)((blockIdx.x * (long long)blockDim.x + threadIdx.x) >> 5);
    int nwaves = (int)(((long long)gridDim.x * blockDim.x) >> 5);
    int lane = threadIdx.x & 31;
    for (int e = gwave; e < ETOT; e += nwaves) {
        int si, di;
        edge_endpoints(ei, e, si, di);
        float alpha = ew[e] / (denom[di] + 1e-16f);
        const float* hs = h + (size_t)si * DHID;
        float* od = out + (size_t)di * DHID;
        atomicAdd(&od[lane], alpha * hs[lane]);
        atomicAdd(&od[lane + 32], alpha * hs[lane + 32]);
    }
}

// Epilogue: out += bias, optional ReLU (in place on d_out region)
__global__ __launch_bounds__(256) void gat_bias_act(
    float* __restrict__ out, const float* __restrict__ b, int n, int do_relu) {
    int i = blockIdx.x * blockDim.x + threadIdx.x;
    if (i >= n * DHID) return;
    float v = out[i] + b[i & (DHID - 1)];
    if (do_relu) v = fmaxf(v, 0.0f);
    out[i] = v;
}

// ---------------------------------------------------------------------------
static void run_gat_layer(const float* x_in, int dk, const float* W,
                          const float* a_src, const float* a_dst,
                          const float* bias, const int* ei, float* out_region,
                          int do_relu, float* h, float* s, float* d, float* m,
                          float* denom, float* logit, hipStream_t stream) {
    const int n = NNODES;
    // 1. h = x @ W  (WMMA)
    {
        int tasks = (n / 16) * (DHID / 16);
        gat_gemm_wmma_f32<<<(tasks + 7) / 8, 256, 0, stream>>>(x_in, W, h, n,
                                                               dk, DHID);
    }
    // 2. attention dots
    gat_att_dots<<<(n + 255) / 256, 256, 0, stream>>>(h, a_src, a_dst, s, d, n);
    // 3. init: m=-inf, denom=0, out_region=0
    fill_f32<<<1024, 256, 0, stream>>>(m, -INFINITY, n);
    fill_f32<<<1024, 256, 0, stream>>>(denom, 0.0f, n);
    fill_f32<<<4096, 256, 0, stream>>>(out_region, 0.0f, (long long)n * DHID);
    // 4-6. edge passes
    gat_edge_logit_max<<<(ETOT + 255) / 256, 256, 0, stream>>>(ei, s, d, logit,
                                                               m);
    gat_edge_exp_sum<<<(ETOT + 255) / 256, 256, 0, stream>>>(ei, m, logit,
                                                             denom);
    gat_edge_aggregate<<<(ETOT + 7) / 8, 256, 0, stream>>>(ei, logit, denom, h,
                                                           out_region);
    // 7. epilogue
    gat_bias_act<<<(n * DHID + 255) / 256, 256, 0, stream>>>(out_region, bias,
                                                             n, do_relu);
}

extern "C" void kernel_launch(void* const* d_in, const int* in_sizes, int n_in,
                              void* d_out, int out_size, void* d_ws,
                              size_t ws_size, hipStream_t stream) {
    (void)in_sizes; (void)n_in; (void)out_size; (void)ws_size;
    const float* x      = (const float*)d_in[0];
    const int*   ei     = (const int*)d_in[1];
    const float* W1     = (const float*)d_in[2];
    const float* asrc1  = (const float*)d_in[3];
    const float* adst1  = (const float*)d_in[4];
    const float* b1     = (const float*)d_in[5];
    const float* W2     = (const float*)d_in[6];
    const float* asrc2  = (const float*)d_in[7];
    const float* adst2  = (const float*)d_in[8];
    const float* b2     = (const float*)d_in[9];

    float* emb  = (float*)d_out;                       // [N, 64]
    float* out2 = (float*)d_out + (size_t)NNODES * DHID;  // [N, 64]

    // Workspace carve-up (floats): h | s | d | m | denom | logit
    float* ws    = (float*)d_ws;
    float* h     = ws;                       ws += (size_t)NNODES * DHID;
    float* s     = ws;                       ws += NNODES;
    float* dd    = ws;                       ws += NNODES;
    float* m     = ws;                       ws += NNODES;
    float* denom = ws;                       ws += NNODES;
    float* logit = ws;                       ws += ETOT;

    // Layer 1: x[N,128] -> emb[N,64] (ReLU)
    run_gat_layer(x, 128, W1, asrc1, adst1, b1, ei, emb, /*relu=*/1, h, s, dd,
                  m, denom, logit, stream);
    // Layer 2: emb[N,64] -> out2[N,64] (no activation)
    run_gat_layer(emb, 64, W2, asrc2, adst2, b2, ei, out2, /*relu=*/0, h, s,
                  dd, m, denom, logit, stream);
}